// NMPPlanningModel_54305566491055
// MI455X (gfx1250) — compile-verified
//
#include <hip/hip_runtime.h>

typedef float v2f __attribute__((ext_vector_type(2)));
typedef float v8f __attribute__((ext_vector_type(8)));

#define BATCHES   8192
#define T_TRAJ    12
#define HW        112
#define MAP_ELEMS (112 * 112)
#define PIXEL_SZ  0.5f
#define BLK       256
#define NBLOCKS   (BATCHES / BLK)   // 32

// Wave32 sum-reduction using the fp32 WMMA pipe.
// A (16x4 f32, 2 VGPRs/lane): lane L<16 -> A[L][0]=v, A[L][1]=0;
//                             lane L>=16 -> A[L-16][2]=v, A[L-16][3]=0.
// B = all-ones (layout invariant) => D[m][n] = v_m + v_{m+16} for all n.
// Each lane's 8 accumulator regs hold 8 rows of D; summing them gives
// rows 0-7 (lanes 0-15) or rows 8-15 (lanes 16-31). Two shuffles finish.
__device__ __forceinline__ float wave_reduce_wmma(float v) {
    v2f a;  a[0] = v;    a[1] = 0.0f;
    v2f bm; bm[0] = 1.0f; bm[1] = 1.0f;
    v8f c = {};
    c = __builtin_amdgcn_wmma_f32_16x16x4_f32(
            /*neg_a=*/false, a, /*neg_b=*/false, bm,
            /*c_mod=*/(short)0, c, /*reuse_a=*/false, /*reuse_b=*/false);
    float s = ((c[0] + c[1]) + (c[2] + c[3])) + ((c[4] + c[5]) + (c[6] + c[7]));
    // total = rows(0..7) partial (lane 0) + rows(8..15) partial (lane 16)
    return __shfl(s, 0, 32) + __shfl(s, 16, 32);
}

__global__ __launch_bounds__(BLK) void nmp_hinge_partials(
        const float* __restrict__ cost_map,
        const int4* __restrict__ gt_idx4,
        const int4* __restrict__ neg_idx4,
        float* __restrict__ partials) {
    const int b = blockIdx.x * BLK + threadIdx.x;

    // 24 ints per batch, contiguous & 16B aligned -> 6 x int4 each
    int4 g4[6], n4[6];
#pragma unroll
    for (int i = 0; i < 6; ++i) {
        g4[i] = gt_idx4[b * 6 + i];
        n4[i] = neg_idx4[b * 6 + i];
    }
    const int* gi = (const int*)g4;
    const int* ni = (const int*)n4;
    const float* map = cost_map + (size_t)b * MAP_ELEMS;

    float dist = 0.0f;
    float diff[T_TRAJ];
#pragma unroll
    for (int t = 0; t < T_TRAJ; ++t) {
        const int gr = gi[2 * t], gc = gi[2 * t + 1];
        const int nr = ni[2 * t], nc = ni[2 * t + 1];
        const float gcost = map[gr * HW + gc];
        const float ncost = map[nr * HW + nc];
        diff[t] = gcost - ncost;
        const float dx = (float)(gr - nr);
        const float dy = (float)(gc - nc);
        dist += sqrtf(dx * dx + dy * dy);
    }

    const float dterm = dist * PIXEL_SZ;
    float hinge = 0.0f;
#pragma unroll
    for (int t = 0; t < T_TRAJ; ++t)
        hinge += fmaxf(diff[t] + dterm, 0.0f);

    // deterministic block reduction: WMMA wave-reduce, then serial over 8 waves
    const float wsum = wave_reduce_wmma(hinge);
    __shared__ float lds[BLK / 32];
    const int lane = threadIdx.x & 31;
    const int wave = threadIdx.x >> 5;
    if (lane == 0) lds[wave] = wsum;
    __syncthreads();
    if (threadIdx.x == 0) {
        float s = 0.0f;
#pragma unroll
        for (int w = 0; w < BLK / 32; ++w) s += lds[w];
        partials[blockIdx.x] = s;
    }
}

// Exactly NBLOCKS(=32) partials: one value per lane of a single wave.
__global__ __launch_bounds__(32) void nmp_final_reduce(
        const float* __restrict__ partials, float* __restrict__ out) {
    const float v = partials[threadIdx.x];
    const float tot = wave_reduce_wmma(v);
    if (threadIdx.x == 0) out[0] = tot * (1.0f / (float)BATCHES);
}

extern "C" void kernel_launch(void* const* d_in, const int* in_sizes, int n_in,
                              void* d_out, int out_size, void* d_ws, size_t ws_size,
                              hipStream_t stream) {
    const float* cost_map = (const float*)d_in[0];
    const int4*  gt_idx4  = (const int4*)d_in[1];
    const int4*  neg_idx4 = (const int4*)d_in[2];
    float* out      = (float*)d_out;
    float* partials = (float*)d_ws;   // NBLOCKS floats of scratch

    nmp_hinge_partials<<<NBLOCKS, BLK, 0, stream>>>(cost_map, gt_idx4, neg_idx4, partials);
    nmp_final_reduce<<<1, 32, 0, stream>>>(partials, out);
}